// CensorNet_12163347382920
// MI455X (gfx1250) — compile-verified
//
#include <hip/hip_runtime.h>
#include <math.h>

typedef float v2f __attribute__((ext_vector_type(2)));
typedef float v4f __attribute__((ext_vector_type(4)));
typedef float v8f __attribute__((ext_vector_type(8)));
typedef int   v4i __attribute__((ext_vector_type(4)));

#define AS1 __attribute__((address_space(1)))
#define AS3 __attribute__((address_space(3)))

#if __has_builtin(__builtin_amdgcn_global_load_async_to_lds_b128) && \
    __has_builtin(__builtin_amdgcn_s_wait_asynccnt)
#define HAVE_ASYNC_LDS 1
#else
#define HAVE_ASYNC_LDS 0
#endif

#define T_STEPS 512
#define BATCH   512
#define HID     128
#define B_TILE  16
#define NTHREADS 768            // 24 waves
#define XS_STRIDE 132           // padded LDS row stride (floats) for x / h tiles
#define XBUF (B_TILE * XS_STRIDE)
#define ST_STRIDE 388           // padded LDS row stride for 384-wide staging
#define SH_OFF (B_TILE * ST_STRIDE)   // offset of h-part inside combined S
#define EPS_NLL 1e-4f

__global__ void censor_zero_out(float* out) { out[0] = 0.0f; }

__global__ __launch_bounds__(NTHREADS, 1)
void censor_gru_scan(const float* __restrict__ x,     // [T,B,I]
                     const float* __restrict__ gt,    // [T,B,1]
                     const float* __restrict__ W_ih,  // [3H,I]
                     const float* __restrict__ W_hh,  // [3H,H]
                     const float* __restrict__ b_ih,  // [3H]
                     const float* __restrict__ b_hh,  // [3H]
                     const float* __restrict__ W_dec, // [1,H]
                     const float* __restrict__ b_dec, // [1]
                     float* __restrict__ out)
{
  __shared__ float xs[2 * XBUF];             // double-buffered x tile
  __shared__ float hs[XBUF];
  __shared__ float S[2 * SH_OFF];            // [0,SH_OFF): x@W_ih^T pre-acts
                                             // [SH_OFF,2*SH_OFF): h@W_hh^T pre-acts
  __shared__ float bias_rz[256];             // b_ih + b_hh for r,z columns
  __shared__ float bias_xn[128];             // b_ih for n columns
  __shared__ float bias_hn[128];             // b_hh for n columns
  __shared__ float wdec_s[128];
  __shared__ float wave_nll[16];

  const int tid   = threadIdx.x;
  const int wave  = tid >> 5;          // 0..23
  const int lane  = tid & 31;
  const int row16 = lane & 15;         // M (A) or N (B/C) index within 16-tile
  const int khalf = lane >> 4;         // which K pair of the 16x16x4 fragment
  const int b0    = blockIdx.x * B_TILE;

  // ---- one-time preloads -------------------------------------------------
  for (int i = tid; i < 256; i += NTHREADS) bias_rz[i] = b_ih[i] + b_hh[i];
  for (int i = tid; i < 128; i += NTHREADS) {
    bias_xn[i] = b_ih[256 + i];
    bias_hn[i] = b_hh[256 + i];
    wdec_s[i]  = W_dec[i];
  }
  for (int i = tid; i < XBUF; i += NTHREADS) hs[i] = 0.0f;  // h0 = 0

  // Register-resident weight fragments, loop-invariant across all 511 steps.
  // Wave w owns gate columns [16w, 16w+16) of BOTH projections.
  // 2 x 32 x v2f = 128 VGPRs of weights per lane; ~160 available/wave at
  // 24 waves/WGP, so everything else must stay lean.
  v2f wih[32], whh[32];
  {
    const int wrow = 16 * wave + row16;                 // 0..383
#pragma unroll
    for (int k = 0; k < 32; ++k) {
      const int col = 4 * k + 2 * khalf;
      wih[k] = *(const v2f*)(W_ih + wrow * HID + col);
      whh[k] = *(const v2f*)(W_hh + wrow * HID + col);
    }
  }
  const float bdec = b_dec[0];
  float nll_local = 0.0f;

  // per-thread loader geometry (single 32-bit offset; per-step base is SGPR)
  const int br   = tid >> 5;                 // 0..15 batch row in tile
  const int c4   = (tid & 31) * 4;           // 0..124
  const int xoff = ((b0 + br) * HID + c4);   // element offset within one slice
  float* xs_wr   = xs + br * XS_STRIDE + c4; // LDS write base (buffer 0)

  // stage x[0] into buffer 0
  if (tid < 512) *(v4f*)xs_wr = *(const v4f*)(x + xoff);
  __syncthreads();

  // ---- sequential scan over time ----------------------------------------
  for (int t = 0; t < T_STEPS - 1; ++t) {
    const float* xcur = xs + (t & 1) * XBUF;
    const float* xnsrc = x + (size_t)(t + 1) * (BATCH * HID) + xoff; // saddr+voff
    float*       xnxt_wr = xs_wr + ((t + 1) & 1) * XBUF;

    // Pipeline: x[t+1] global->LDS, overlapped with the WMMA phase.
#if HAVE_ASYNC_LDS
    if (tid < 512) {
      __builtin_amdgcn_global_load_async_to_lds_b128(
          (AS1 v4i*)xnsrc, (AS3 v4i*)xnxt_wr, 0, 0);
      __builtin_prefetch(xnsrc + (size_t)(BATCH * HID), 0, 1);  // warm GL2 (speculative)
    }
#else
    v4f xnext;
    if (tid < 512) {
      xnext = *(const v4f*)xnsrc;
      __builtin_prefetch(xnsrc + (size_t)(BATCH * HID), 0, 1);
    }
#endif

    // Phase B: two independent WMMA chains per wave.
    v8f acc_x = {0.f,0.f,0.f,0.f,0.f,0.f,0.f,0.f};
    v8f acc_h = acc_x;
    {
      const float* ax_base = xcur + row16 * XS_STRIDE;
      const float* ah_base = hs   + row16 * XS_STRIDE;
#pragma unroll
      for (int half = 0; half < 2; ++half) {
        v2f axf[16], ahf[16];
#pragma unroll
        for (int k = 0; k < 16; ++k) {
          const int col = 4 * (16 * half + k) + 2 * khalf;
          axf[k] = *(const v2f*)(ax_base + col);
          ahf[k] = *(const v2f*)(ah_base + col);
        }
#pragma unroll
        for (int k = 0; k < 16; ++k) {
          acc_x = __builtin_amdgcn_wmma_f32_16x16x4_f32(
                      false, axf[k], false, wih[16 * half + k], (short)0,
                      acc_x, false, false);
          acc_h = __builtin_amdgcn_wmma_f32_16x16x4_f32(
                      false, ahf[k], false, whh[16 * half + k], (short)0,
                      acc_h, false, false);
        }
      }
    }
    // spill accumulators to LDS staging (C layout: M = i + 8*khalf, N = row16)
    // one in-bounds base into the combined S array covers both halves.
    {
      float* sx = S + 16 * wave + row16;   // column base, x-part
#pragma unroll
      for (int i = 0; i < 8; ++i) {
        const int m = i + 8 * khalf;
        sx[m * ST_STRIDE]          = acc_x[i];
        sx[m * ST_STRIDE + SH_OFF] = acc_h[i];
      }
    }
#if HAVE_ASYNC_LDS
    __builtin_amdgcn_s_wait_asynccnt(0);
#else
    if (tid < 512) *(v4f*)xnxt_wr = xnext;
#endif
    __syncthreads();

    // Phase C: gate nonlinearities + in-place h update (2048 elems)
    for (int idx = tid; idx < B_TILE * HID; idx += NTHREADS) {
      const int b = idx >> 7;
      const int j = idx & 127;
      const float* sxb = S + b * ST_STRIDE;       // x-part row
      const float* shb = sxb + SH_OFF;            // h-part row (in-bounds of S)
      const float r = 1.0f / (1.0f + __expf(-(sxb[      j] + shb[      j] + bias_rz[      j])));
      const float z = 1.0f / (1.0f + __expf(-(sxb[128 + j] + shb[128 + j] + bias_rz[128 + j])));
      const float n = tanhf(sxb[256 + j] + bias_xn[j] + r * (shb[256 + j] + bias_hn[j]));
      const float hv = hs[b * XS_STRIDE + j];
      hs[b * XS_STRIDE + j] = (1.0f - z) * n + z * hv;
    }
    __syncthreads();

    // Phase D: decoder + NLL.  Wave w (w<16) reduces batch row w.
    if (wave < 16) {
      const float* hrow = hs + wave * XS_STRIDE;
      float part = 0.0f;
#pragma unroll
      for (int c = 0; c < 4; ++c) {
        const int j = lane * 4 + c;
        part += hrow[j] * wdec_s[j];
      }
      part += __shfl_xor(part, 16);
      part += __shfl_xor(part, 8);
      part += __shfl_xor(part, 4);
      part += __shfl_xor(part, 2);
      part += __shfl_xor(part, 1);
      if (lane == 0) {
        const float Cv = 1.0f / (1.0f + __expf(-(part + bdec)));
        const float g  = gt[(size_t)(t + 1) * BATCH + b0 + wave];
        nll_local -= g * __logf(Cv + EPS_NLL) + (1.0f - g) * __logf(1.0f - Cv + EPS_NLL);
      }
    }
    // no barrier needed here: next writers of hs are fenced by the next
    // iteration's phase-B barrier; xs buffers alternate.
  }

  // ---- final reduction ---------------------------------------------------
  if (wave < 16 && lane == 0) wave_nll[wave] = nll_local;
  __syncthreads();
  if (tid == 0) {
    float s = 0.0f;
    for (int i = 0; i < 16; ++i) s += wave_nll[i];
    atomicAdd(out, s * (1.0f / (float)(T_STEPS * BATCH)));
  }
}

extern "C" void kernel_launch(void* const* d_in, const int* in_sizes, int n_in,
                              void* d_out, int out_size, void* d_ws, size_t ws_size,
                              hipStream_t stream) {
  (void)in_sizes; (void)n_in; (void)out_size; (void)d_ws; (void)ws_size;
  const float* x     = (const float*)d_in[0];
  const float* gt    = (const float*)d_in[1];
  const float* W_ih  = (const float*)d_in[2];
  const float* W_hh  = (const float*)d_in[3];
  const float* b_ih  = (const float*)d_in[4];
  const float* b_hh  = (const float*)d_in[5];
  const float* W_dec = (const float*)d_in[6];
  const float* b_dec = (const float*)d_in[7];
  float* out = (float*)d_out;

  censor_zero_out<<<1, 1, 0, stream>>>(out);
  censor_gru_scan<<<BATCH / B_TILE, NTHREADS, 0, stream>>>(
      x, gt, W_ih, W_hh, b_ih, b_hh, W_dec, b_dec, out);
}